// VariableSelectionNetwork_41360535061209
// MI455X (gfx1250) — compile-verified
//
#include <hip/hip_runtime.h>
#include <hip/hip_bf16.h>

// Problem constants from the reference.
#define BT   32768   // B*T = 16*2048
#define FF   32      // F
#define DD   128     // D
#define XS_STRIDE 34 // padded LDS stride (dwords): 34*lm mod 64 distinct -> bank-conflict free

typedef __attribute__((ext_vector_type(2))) float v2f;
typedef __attribute__((ext_vector_type(8))) float v8f;

// z[b,t,d] = sum_f x[b,t,f]*Wp[f,d] + sum_f bp[f,d]   (w == 1 because softmax over a
// singleton axis is identically 1, so the whole GRN branch is dead w.r.t. outputs).
// One workgroup (8 wave32s) computes a 16(M) x 128(N) strip of z using
// V_WMMA_F32_16X16X4_F32 (8 K-steps of 4 over K=32), plus the 16x32 ones-chunk of w.
__global__ __launch_bounds__(256) void vsn_gemm_wmma_f32(
    const float* __restrict__ x,    // (BT, FF)
    const float* __restrict__ Wp,   // (FF, DD)
    const float* __restrict__ bp,   // (FF, DD)
    float* __restrict__ z,          // (BT, DD)
    float* __restrict__ w)          // (BT, FF) -> all ones
{
    __shared__ float xs[16 * XS_STRIDE];

    const int tid  = threadIdx.x;        // 0..255
    const int wave = tid >> 5;           // 0..7  -> N tile index
    const int lane = tid & 31;
    const int row0 = blockIdx.x * 16;    // M base of this workgroup

    // Warm the 16KB Wp matrix into cache (global_prefetch_b8). 256 lanes x 64 floats
    // of stride covers the whole 32x128 fp32 matrix region.
    __builtin_prefetch(Wp + tid * 64, 0, 1);

    // Cooperative load of the 16x32 x-tile into padded LDS: thread tid handles
    // row = tid/16, cols = 2*(tid%16)..+1  (float2, 8B aligned since stride is even).
    {
        const int r = tid >> 4;          // 0..15
        const int c = (tid & 15) << 1;   // 0,2,..,30
        const float2 v = *reinterpret_cast<const float2*>(x + (size_t)(row0 + r) * FF + c);
        *reinterpret_cast<float2*>(&xs[r * XS_STRIDE + c]) = v;
    }

    // w == 1.0 for this strip: 16 rows x 32 features = 512 values, 2 per thread.
    {
        const int i = tid << 1;
        float2 ones; ones.x = 1.0f; ones.y = 1.0f;
        *reinterpret_cast<float2*>(w + (size_t)row0 * FF + i) = ones;
    }

    __syncthreads();

    const int half = lane >> 4;          // 0: K pair {0,1}; 1: K pair {2,3} (per K-block)
    const int lm   = lane & 15;
    const int n    = wave * 16 + lm;     // output column for this lane

    // Column bias = sum_f bp[f, n]  (coalesced across lanes, L2-hot 16KB).
    float bias = 0.0f;
#pragma unroll
    for (int f = 0; f < FF; ++f) bias += bp[f * DD + n];

    v8f acc = {};
#pragma unroll
    for (int kb = 0; kb < 8; ++kb) {     // K = 32 in steps of 4
        const int k0 = kb * 4 + half * 2;
        v2f a, b;
        // A 16x4 f32: lanes 0-15 M=lm hold K={k0,k0+1} with half=0; lanes 16-31 hold K={+2,+3}
        a.x = xs[lm * XS_STRIDE + k0];
        a.y = xs[lm * XS_STRIDE + k0 + 1];
        // B 4x16 f32: VGPR j, lanes 0-15 -> K=kb*4+j, lanes 16-31 -> K=kb*4+j+2 (mirrors C layout)
        b.x = Wp[(k0)     * DD + n];
        b.y = Wp[(k0 + 1) * DD + n];
        // 8 args: (neg_a, A, neg_b, B, c_mod, C, reuse_a, reuse_b)
        acc = __builtin_amdgcn_wmma_f32_16x16x4_f32(false, a, false, b,
                                                    (short)0, acc, false, false);
    }

    // C/D 16x16 f32 layout: VGPR i, lanes 0-15 -> M=i, lanes 16-31 -> M=i+8; N=lm.
#pragma unroll
    for (int i = 0; i < 8; ++i) {
        const int m = row0 + i + half * 8;
        z[(size_t)m * DD + n] = acc[i] + bias;
    }
}

extern "C" void kernel_launch(void* const* d_in, const int* in_sizes, int n_in,
                              void* d_out, int out_size, void* d_ws, size_t ws_size,
                              hipStream_t stream) {
    // setup_inputs order: x, Wp, bp, W1, b1, W2, b2, Wg, bg, gamma, beta
    const float* x  = (const float*)d_in[0];
    const float* Wp = (const float*)d_in[1];
    const float* bp = (const float*)d_in[2];
    // Outputs concatenated flat in return order: z (BT*DD), then w (BT*FF).
    float* z = (float*)d_out;
    float* w = (float*)d_out + (size_t)BT * DD;

    (void)in_sizes; (void)n_in; (void)out_size; (void)d_ws; (void)ws_size;

    dim3 grid(BT / 16);   // 2048 workgroups
    dim3 block(256);      // 8 wave32s
    hipLaunchKernelGGL(vsn_gemm_wmma_f32, grid, block, 0, stream, x, Wp, bp, z, w);
}